// VectorFieldModule_61804579389705
// MI455X (gfx1250) — compile-verified
//
#include <hip/hip_runtime.h>

typedef float v2f __attribute__((ext_vector_type(2)));
typedef float v8f __attribute__((ext_vector_type(8)));

#define NDIM 192
#define NN   (192 * 192)
#define VOL  (192 * 192 * 192)
#define RAD  6
#define KS   13

// ---------------------------------------------------------------------------
// Recover exact 1D Gaussian factors from the 3D weight via marginal sums.
// w layout: (3, 1, 13, 13, 13). Since w[c] = g_i ⊗ g_j ⊗ g_k with each factor
// summing to 1, summing over the other two axes returns the factor exactly.
// gw layout: [axis(3)][channel(3)][tap(13)].
// ---------------------------------------------------------------------------
__global__ void factor_kernel(const float* __restrict__ w, float* __restrict__ gw) {
  int t = threadIdx.x;
  if (t >= 3 * 3 * KS) return;
  int axis = t / (3 * KS);
  int c    = (t % (3 * KS)) / KS;
  int tap  = t % KS;
  const float* wc = w + (size_t)c * (KS * KS * KS);
  float s = 0.f;
  for (int a = 0; a < KS; ++a) {
    for (int b = 0; b < KS; ++b) {
      int i, j, k;
      if (axis == 0)      { i = tap; j = a;   k = b; }
      else if (axis == 1) { i = a;   j = tap; k = b; }
      else                { i = a;   j = b;   k = tap; }
      s += wc[(i * KS + j) * KS + k];
    }
  }
  gw[t] = s;
}

__device__ __forceinline__ int clampi(int v, int lo, int hi) {
  return v < lo ? lo : (v > hi ? hi : v);
}

// ---------------------------------------------------------------------------
// One separable 1D conv pass via V_WMMA_F32_16X16X4_F32.
// Each wave owns one group of 16 lines and sweeps all 12 position tiles along
// the conv axis: per tile, D(16x16) = W_band(16x28) * X(28x16) in 7 K=4 chunks.
// Strides are template constants so address math is immediate/shift only and
// adjacent accesses merge (b64 loads / b128 stores on the contiguous pass).
// Edge chunks use clamp-address + cndmask-to-zero (branch-free; EXEC stays
// all-ones as WMMA requires); chunks entirely outside [0,192) are skipped at
// unroll time.
// A 16x4 f32 layout (ISA 7.12.2): elem A[m][k] -> lane = m + 16*(k>=2), reg = k&1.
// B  4x16 f32 layout:             elem B[k][n] -> lane = n + 16*(k>=2), reg = k&1.
// D 16x16 f32 layout:             reg v, lane L -> row = v + 8*(L>=16), col = L&15.
// ---------------------------------------------------------------------------
template <int SC, int SL, int SO>
__global__ __launch_bounds__(256) void conv1d_wmma(
    const float* __restrict__ src, float* __restrict__ dst,
    const float* __restrict__ g13 /* [3][13] taps for this axis */) {
  int wave = blockIdx.x * 8 + (threadIdx.x >> 5);
  int lane = threadIdx.x & 31;
  int m = lane & 15;  // A row / B col / D col held by this lane
  int h = lane >> 4;  // lane half: selects K sub-pair and D row offset

  // wave -> (line group, outer line dim, channel); 12 * 192 * 3 = 6912 waves.
  int jg    = wave % 12;
  int outer = (wave / 12) % NDIM;
  int c     = wave / (12 * NDIM);

  // Banded weight chunks: W[m][k] = g[k - m] for 0 <= k-m <= 12.
  // Clamp-load + mask instead of register-indexed gather (no select trees).
  const float* gc = g13 + c * KS;
  v2f A[7];
#pragma unroll
  for (int q = 0; q < 7; ++q) {
    int t0 = 4 * q + 2 * h - m;
    int t1 = t0 + 1;
    float a0 = gc[clampi(t0, 0, KS - 1)];
    float a1 = gc[clampi(t1, 0, KS - 1)];
    A[q].x = (t0 >= 0 && t0 < KS) ? a0 : 0.f;
    A[q].y = (t1 >= 0 && t1 < KS) ? a1 : 0.f;
  }

  size_t base = (size_t)c * VOL + (size_t)outer * SO + (size_t)(jg * 16 + m) * SL;
  const float* sp = src + base;
  float* dp = dst + base;

#pragma unroll
  for (int pt = 0; pt < 12; ++pt) {
    v8f acc = {};
#pragma unroll
    for (int q = 0; q < 7; ++q) {
      const int pbase = pt * 16 - RAD + 4 * q;  // unroll-time constant
      // Chunk covers positions [pbase, pbase+3]; skip if fully out of range
      // (its B block is identically zero).
      if (pbase + 3 < 0 || pbase > NDIM - 1) continue;
      float bx, by;
      if (pbase >= 0 && pbase + 3 < NDIM) {
        // Interior chunk: unconditional loads, zero boundary logic.
        int p = pbase + 2 * h;
        bx = sp[(size_t)p * SC];
        by = sp[(size_t)(p + 1) * SC];
      } else {
        // Edge chunk: clamp address (always valid), mask value to zero.
        int p  = pbase + 2 * h;
        int p1 = p + 1;
        bx = sp[(size_t)clampi(p,  0, NDIM - 1) * SC];
        by = sp[(size_t)clampi(p1, 0, NDIM - 1) * SC];
        bx = (p  >= 0 && p  < NDIM) ? bx : 0.f;
        by = (p1 >= 0 && p1 < NDIM) ? by : 0.f;
      }
      v2f B = {bx, by};
      acc = __builtin_amdgcn_wmma_f32_16x16x4_f32(
          /*neg_a=*/false, A[q], /*neg_b=*/false, B,
          /*c_mod=*/(short)0, acc, /*reuse_a=*/false, /*reuse_b=*/false);
    }
#pragma unroll
    for (int v = 0; v < 8; ++v) {
      dp[(size_t)(pt * 16 + v + 8 * h) * SC] = acc[v];
    }
  }
}

// ---------------------------------------------------------------------------
// Deterministic fallback if workspace is too small: direct 13^3 depthwise conv.
// ---------------------------------------------------------------------------
__global__ void conv3d_direct(const float* __restrict__ x,
                              const float* __restrict__ w,
                              float* __restrict__ out, int total) {
  for (long idx = (long)blockIdx.x * blockDim.x + threadIdx.x; idx < total;
       idx += (long)gridDim.x * blockDim.x) {
    int z  = (int)(idx % NDIM);
    int y  = (int)((idx / NDIM) % NDIM);
    int xx = (int)((idx / NN) % NDIM);
    int c  = (int)(idx / VOL);
    const float* xc = x + (size_t)c * VOL;
    const float* wc = w + (size_t)c * (KS * KS * KS);
    float s = 0.f;
    for (int i = 0; i < KS; ++i) {
      int px = xx + i - RAD;
      if (px < 0 || px >= NDIM) continue;
      for (int j = 0; j < KS; ++j) {
        int py = y + j - RAD;
        if (py < 0 || py >= NDIM) continue;
        const float* row  = xc + (size_t)px * NN + (size_t)py * NDIM;
        const float* wrow = wc + (i * KS + j) * KS;
        for (int k = 0; k < KS; ++k) {
          int pz = z + k - RAD;
          if (pz < 0 || pz >= NDIM) continue;
          s += wrow[k] * row[pz];
        }
      }
    }
    out[idx] = s;
  }
}

extern "C" void kernel_launch(void* const* d_in, const int* in_sizes, int n_in,
                              void* d_out, int out_size, void* d_ws, size_t ws_size,
                              hipStream_t stream) {
  const float* x = (const float*)d_in[0];  // (1,3,192,192,192) f32
  const float* w = (const float*)d_in[1];  // (3,1,13,13,13) f32
  float* out = (float*)d_out;

  size_t need = (size_t)3 * VOL * sizeof(float) + 3 * 3 * KS * sizeof(float);
  if (ws_size >= need) {
    float* tmp = (float*)d_ws;
    float* gw  = tmp + (size_t)3 * VOL;

    factor_kernel<<<1, 128, 0, stream>>>(w, gw);

    int waves  = 3 * 12 * NDIM;  // 6912 line-group waves, 12 tiles each
    int blocks = waves / 8;      // 8 waves per 256-thread block, exact

    // Pass 1: conv along z (contiguous). lines: j = y (stride N), outer = x (stride NN).
    conv1d_wmma<1, NDIM, NN><<<blocks, 256, 0, stream>>>(x, out, gw + 2 * 3 * KS);
    // Pass 2: conv along y (stride N). lines: j = z (stride 1), outer = x (stride NN).
    conv1d_wmma<NDIM, 1, NN><<<blocks, 256, 0, stream>>>(out, tmp, gw + 1 * 3 * KS);
    // Pass 3: conv along x (stride NN). lines: j = z (stride 1), outer = y (stride N).
    conv1d_wmma<NN, 1, NDIM><<<blocks, 256, 0, stream>>>(tmp, out, gw + 0 * 3 * KS);
  } else {
    conv3d_direct<<<4096, 256, 0, stream>>>(x, w, out, 3 * VOL);
  }
}